// WeightsLib2DMobius_47691316855288
// MI455X (gfx1250) — compile-verified
//
#include <hip/hip_runtime.h>
#include <hip/hip_bf16.h>
#include <math.h>

typedef __attribute__((ext_vector_type(2))) float v2f;
typedef __attribute__((ext_vector_type(8))) float v8f;

#define NB   2      // batch
#define CTXN 512
#define TT   34     // 2 + 2*RT
#define CC   16     // C
#define RSN  16     // RS
#define LLN  128    // L
#define KKN  11     // K
#define SZN  1024   // H == W
#define RTN  16

// -------------------------------------------------------------------------
// Kernel 1: xt = x @ W_ctx + b_ctx  via V_WMMA_F32_16X16X4_F32 (fp32 exact).
// One wave (32 threads) per 16-column output tile; M=2 padded to 16 with 0s.
// A-frag layout (ISA 7.12.2, 32-bit A 16x4): lane m=lane&15, kpair=(lane>>4)*2,
//   a.x=A[m][kpair], a.y=A[m][kpair+1].  B-frag symmetric (4x16, K striped).
// D layout: VGPR v holds M=v (lanes 0-15) / M=v+8 (lanes 16-31), N=lane&15.
// -------------------------------------------------------------------------
__global__ __launch_bounds__(32) void gemm_ctx_kernel(
    const float* __restrict__ x, const float* __restrict__ Wc,
    const float* __restrict__ bias, float* __restrict__ xt) {
  const int lane = threadIdx.x;
  const int n0   = blockIdx.x * 16;
  const int m    = lane & 15;
  const int kk   = (lane >> 4) * 2;     // 0 or 2
  const int row  = (m < NB) ? m : 0;    // clamp to avoid OOB read; masked below
  v8f acc = {};
  for (int k0 = 0; k0 < CTXN; k0 += 4) {
    v2f a, b;
    float a0 = x[row * CTXN + k0 + kk];
    float a1 = x[row * CTXN + k0 + kk + 1];
    a.x = (m < NB) ? a0 : 0.0f;         // pad M rows 2..15 with zeros
    a.y = (m < NB) ? a1 : 0.0f;
    b.x = Wc[(size_t)(k0 + kk) * 512 + n0 + m];
    b.y = Wc[(size_t)(k0 + kk + 1) * 512 + n0 + m];
    acc = __builtin_amdgcn_wmma_f32_16x16x4_f32(
        /*neg_a=*/false, a, /*neg_b=*/false, b,
        /*c_mod=*/(short)0, acc, /*reuse_a=*/false, /*reuse_b=*/false);
  }
  if (lane < 16) {
    float bv = bias[n0 + lane];
    xt[0 * 512 + n0 + lane] = acc[0] + bv;   // M=0
    xt[1 * 512 + n0 + lane] = acc[1] + bv;   // M=1
  }
}

// -------------------------------------------------------------------------
// Kernel 2: siglog gate + complex square.  i = b*128 + l.
// -------------------------------------------------------------------------
__global__ __launch_bounds__(256) void x2_kernel(const float* __restrict__ xt,
                                                 float* __restrict__ x2) {
  const int i = threadIdx.x;            // 0..255
  const int b = i >> 7, l = i & 127;
  const float* rowp = xt + b * 512;
  float pr = rowp[2 * l],      pi = rowp[2 * l + 1];
  float qr = rowp[256 + 2 * l], qi = rowp[256 + 2 * l + 1];
  float xr = pr * copysignf(log1pf(fabsf(qr)), qr);
  float xi = pi * copysignf(log1pf(fabsf(qi)), qi);
  x2[i * 2 + 0] = xr * xr - xi * xi;    // x2r
  x2[i * 2 + 1] = 2.0f * xr * xi;       // x2i
}

// -------------------------------------------------------------------------
// Kernel 3: dominant streaming stage. One block per (t,c); 256 threads.
// Thread: r = tid&15, l = (tid>>4) + 16*j for j=0..7.
//   AB[l,r]   = sum_k A[l,k,r] * B[k,l,r]           (complex, K=11)
//   for b: Re/Im = x2[b,l] * AB[l,r];  mag = sqrt(Re^2+Im^2+eps)
//   reduce over l (128) -> tr[b,t,c,r,{0,1}]
// All A/B traffic is coalesced float2 (re,im pairs contiguous in memory).
// -------------------------------------------------------------------------
__global__ __launch_bounds__(256) void mobius_main_kernel(
    const float* __restrict__ A, const float* __restrict__ Bf,
    const float* __restrict__ x2, float* __restrict__ tr) {
  __shared__ float s_x2[512];
  __shared__ float s_red[6 * 256];
  const int tid = threadIdx.x;
  const int t_idx = blockIdx.x / CC;
  const int c_idx = blockIdx.x % CC;

  s_x2[tid]       = x2[tid];
  s_x2[tid + 256] = x2[tid + 256];
  __syncthreads();

  const v2f* A2 = (const v2f*)A + (size_t)blockIdx.x * (LLN * KKN * RSN);
  const v2f* B2 = (const v2f*)Bf + (size_t)blockIdx.x * (LLN * KKN * RSN);

  const int r  = tid & 15;
  const int lb = tid >> 4;
  float sRe0 = 0.f, sIm0 = 0.f, sMg0 = 0.f;
  float sRe1 = 0.f, sIm1 = 0.f, sMg1 = 0.f;

  #pragma unroll 2
  for (int j = 0; j < 8; ++j) {
    const int l = lb + 16 * j;
    // hint the next l-stripe into cache (speculative; dropped if OOB)
    __builtin_prefetch(&A2[(size_t)(l + 16) * (KKN * RSN) + r], 0, 1);
    __builtin_prefetch(&B2[(size_t)(l + 16) * RSN + r], 0, 1);
    float abr = 0.f, abi = 0.f;
    #pragma unroll
    for (int k = 0; k < KKN; ++k) {
      v2f a = A2[(size_t)l * (KKN * RSN) + k * RSN + r];
      v2f b = B2[(size_t)k * (LLN * RSN) + l * RSN + r];
      abr = fmaf(a.x, b.x, fmaf(-a.y, b.y, abr));
      abi = fmaf(a.x, b.y, fmaf(a.y, b.x, abi));
    }
    {
      float xr2 = s_x2[l * 2], xi2 = s_x2[l * 2 + 1];
      float Re = xr2 * abr - xi2 * abi;
      float Im = xr2 * abi + xi2 * abr;
      float mg = sqrtf(fmaf(Re, Re, fmaf(Im, Im, 1e-12f)));
      sRe0 += Re; sIm0 += Im; sMg0 += mg;
    }
    {
      float xr2 = s_x2[256 + l * 2], xi2 = s_x2[256 + l * 2 + 1];
      float Re = xr2 * abr - xi2 * abi;
      float Im = xr2 * abi + xi2 * abr;
      float mg = sqrtf(fmaf(Re, Re, fmaf(Im, Im, 1e-12f)));
      sRe1 += Re; sIm1 += Im; sMg1 += mg;
    }
  }

  s_red[tid]            = sRe0;
  s_red[256 + tid]      = sIm0;
  s_red[512 + tid]      = sMg0;
  s_red[768 + tid]      = sRe1;
  s_red[1024 + tid]     = sIm1;
  s_red[1280 + tid]     = sMg1;
  __syncthreads();

  if (tid < 16) {       // one thread per r; sum the 16 partials sharing r
    float v0 = 0.f, v1 = 0.f, v2 = 0.f, v3 = 0.f, v4 = 0.f, v5 = 0.f;
    #pragma unroll
    for (int g = 0; g < 16; ++g) {
      int base = tid + 16 * g;
      v0 += s_red[base];        v1 += s_red[256 + base];  v2 += s_red[512 + base];
      v3 += s_red[768 + base];  v4 += s_red[1024 + base]; v5 += s_red[1280 + base];
    }
    // cosw*mag_mean = (sumRe/(msum+eps)) * (msum/L)
    {
      float f = (v2 * (1.0f / 128.0f)) / (v2 + 1e-12f);
      size_t o = ((((size_t)0 * TT + t_idx) * CC + c_idx) * RSN + tid) * 2;
      tr[o]     = v0 * f;
      tr[o + 1] = v1 * f;
    }
    {
      float f = (v5 * (1.0f / 128.0f)) / (v5 + 1e-12f);
      size_t o = ((((size_t)1 * TT + t_idx) * CC + c_idx) * RSN + tid) * 2;
      tr[o]     = v3 * f;
      tr[o + 1] = v4 * f;
    }
  }
}

// -------------------------------------------------------------------------
// Kernel 4: linear interp of tr along RT->1024 + "space" Mobius combine.
// One thread per output float2: g -> (out_sel, b, c, rs, h).
// -------------------------------------------------------------------------
__global__ __launch_bounds__(256) void interp_kernel(
    const float* __restrict__ space_i, const float* __restrict__ space_j,
    const float* __restrict__ tr, float* __restrict__ out) {
  const int g = blockIdx.x * 256 + threadIdx.x;
  const int perOut = NB * CC * RSN * SZN;        // 524288 float2 per output
  const int os  = g / perOut;
  int rem = g - os * perOut;
  const int b  = rem / (CC * RSN * SZN); rem -= b * (CC * RSN * SZN);
  const int c  = rem / (RSN * SZN);      rem -= c * (RSN * SZN);
  const int rs = rem / SZN;
  const int h  = rem - rs * SZN;

  const v2f* sp2 = (const v2f*)(os ? space_j : space_i);
  v2f s = sp2[(size_t)(c * RSN + rs) * SZN + h];

  const v2f* tr2 = (const v2f*)tr;
  v2f sh = tr2[(((size_t)b * TT + os) * CC + c) * RSN + rs];   // shift = tr[:,os]

  // F.interpolate(linear, align_corners=False): 16 -> 1024
  float pos = (h + 0.5f) * (16.0f / 1024.0f) - 0.5f;
  pos = fminf(fmaxf(pos, 0.0f), 15.0f);
  int   i0 = (int)pos;
  int   i1 = (i0 + 1 > 15) ? 15 : (i0 + 1);
  float w  = pos - (float)i0;

  size_t zbase = (((size_t)b * TT + (2 + os * RTN)) * CC + c) * RSN + rs;
  v2f za = tr2[zbase + (size_t)i0 * (CC * RSN)];
  v2f zb = tr2[zbase + (size_t)i1 * (CC * RSN)];
  float z0 = za.x * (1.0f - w) + zb.x * w;
  float z1 = za.y * (1.0f - w) + zb.y * w;

  float sp0 = s.x + sh.x, sp1 = s.y + sh.y;
  float re = sp1 * z1 - sp0 * z0;          // prod[...,1] - prod[...,0]
  float im = sp0 * z1 + sp1 * z0;          // sp . reverse(z)

  v2f* o2 = (v2f*)out;
  v2f res; res.x = re; res.y = im;
  o2[(size_t)os * perOut + ((size_t)(b * CC + c) * RSN + rs) * SZN + h] = res;
}

// -------------------------------------------------------------------------
extern "C" void kernel_launch(void* const* d_in, const int* in_sizes, int n_in,
                              void* d_out, int out_size, void* d_ws, size_t ws_size,
                              hipStream_t stream) {
  const float* x       = (const float*)d_in[0];
  const float* Wc      = (const float*)d_in[1];
  const float* bias    = (const float*)d_in[2];
  const float* A       = (const float*)d_in[3];
  const float* Bf      = (const float*)d_in[4];
  const float* space_i = (const float*)d_in[5];
  const float* space_j = (const float*)d_in[6];

  float* ws = (float*)d_ws;
  float* xt = ws;            // 1024 floats
  float* x2 = ws + 1024;     // 512 floats
  float* tr = ws + 1536;     // 34816 floats (B,T,C,RS,2)
  float* out = (float*)d_out;

  gemm_ctx_kernel<<<32, 32, 0, stream>>>(x, Wc, bias, xt);
  x2_kernel<<<1, 256, 0, stream>>>(xt, x2);
  mobius_main_kernel<<<TT * CC, 256, 0, stream>>>(A, Bf, x2, tr);
  const int nThreads = 2 * NB * CC * RSN * SZN;   // 1,048,576 float2 outputs
  interp_kernel<<<nThreads / 256, 256, 0, stream>>>(space_i, space_j, tr, out);
}